// AngularSymmetryMod_68315749810415
// MI455X (gfx1250) — compile-verified
//
#include <hip/hip_runtime.h>
#include <hip/hip_bf16.h>

typedef float v2f __attribute__((ext_vector_type(2)));
typedef float v8f __attribute__((ext_vector_type(8)));

#define BOHR_INV (1.0f / 0.52917721092f)

__global__ __launch_bounds__(128) void AngularSymmetry_gfx1250_kernel(
    const float* __restrict__ d_cutoff,   // (B, N, N)
    const float* __restrict__ d,          // (B, N, N)
    const float* __restrict__ coords,     // (B, N, 3)
    float* __restrict__ out)              // (B, N, 40)
{
    constexpr int N = 32;
    constexpr int NP = 40;   // 2 lambda * 5 Rs * 4 thetas
    constexpr int NT = 128;  // threads per block (4 waves)

    const int bi   = blockIdx.x;      // b*N + i
    const int b    = bi >> 5;
    const int i    = bi & 31;
    const int tid  = threadIdx.x;
    const int wave = tid >> 5;
    const int lane = tid & 31;
    const int half = lane >> 4;       // lane half selects K-pair / M+8
    const int m    = lane & 15;
    const int j0   = (wave >> 1) << 4;
    const int k0   = (wave & 1) << 4;

    __shared__ float sd[N];           // d[b, i, :]
    __shared__ float sf[N];           // d_cutoff[b, i, :]
    __shared__ float red[NP * NT];    // 20 KB reduction scratch

    if (tid < N) {
        sd[tid] = d[bi * N + tid];
        sf[tid] = d_cutoff[bi * N + tid];
    }

    // center atom (uniform across block)
    const float cx = coords[(b * N + i) * 3 + 0];
    const float cy = coords[(b * N + i) * 3 + 1];
    const float cz = coords[(b * N + i) * 3 + 2];

    // A operand row j = j0+m (vec_j), B operand col k = k0+m (vec_k)
    const int j = j0 + m;
    const int k = k0 + m;
    const float jx = coords[(b * N + j) * 3 + 0] - cx;
    const float jy = coords[(b * N + j) * 3 + 1] - cy;
    const float jz = coords[(b * N + j) * 3 + 2] - cz;
    const float kx = coords[(b * N + k) * 3 + 0] - cx;
    const float ky = coords[(b * N + k) * 3 + 1] - cy;
    const float kz = coords[(b * N + k) * 3 + 2] - cz;

    // 16x4 f32 A layout: VGPR v, lane-half h -> K = v + 2h  (K=3 is zero pad)
    v2f A, Bv;
    A[0]  = half ? jz : jx;   // K = 0 or 2
    A[1]  = half ? 0.f : jy;  // K = 1 or 3(pad)
    Bv[0] = half ? kz : kx;
    Bv[1] = half ? 0.f : ky;

    v8f C = {};
    // dot tile (16x16): D = A(16x4) x B(4x16), f32 accumulate
    C = __builtin_amdgcn_wmma_f32_16x16x4_f32(
            /*neg_a=*/false, A, /*neg_b=*/false, Bv,
            /*c_mod=*/(short)0, C, /*reuse_a=*/false, /*reuse_b=*/false);

    __syncthreads();

    // symmetry params (flat order: lambda(2) x Rs(5) x thetas(4))
    const float ita = 1.12f;
    const float Rs[5] = {0.5f * BOHR_INV, 1.17f * BOHR_INV, 1.83f * BOHR_INV,
                         2.5f * BOHR_INV, 3.17f * BOHR_INV};
    // cos/sin of thetas {0, 1.57, 3.14, 4.71}
    const float thc[4] = { 1.0f, 0.00079632671f, -0.99999873172f, -0.00238897813f };
    const float ths[4] = { 0.0f, 0.99999968293f,  0.00159265292f, -0.99999714670f };

    float acc[NP];
#pragma unroll
    for (int p = 0; p < NP; ++p) acc[p] = 0.0f;

    // C layout: VGPR r, lanes 0-15 -> M=r, lanes 16-31 -> M=r+8; N = lane%16
#pragma unroll
    for (int r = 0; r < 8; ++r) {
        const int jj = j0 + r + 8 * half;
        const int kk = k0 + m;
        const float dotv = C[r];

        const float Rij = sd[jj];
        const float Rik = sd[kk];
        const float fij = sf[jj];
        const float fik = sf[kk];

        // NOTE: reference uses cos-similarity value directly as "theta".
        // Denominator is fuzzed (+1e-5) in the reference, so a raw v_rcp_f32
        // is the right throughput-bound choice (avoids div_scale sequence).
        const float th = dotv * __builtin_amdgcn_rcpf(Rij * Rik + 1e-5f);
        const float c  = __cosf(th);
        const float s  = __sinf(th);

        float ct[4];
#pragma unroll
        for (int t = 0; t < 4; ++t) ct[t] = c * thc[t] + s * ths[t];

        float angP[4], angM[4];   // (1 + lambda*ct)^4, lambda = +1 / -1
#pragma unroll
        for (int t = 0; t < 4; ++t) {
            float u = 1.0f + ct[t]; float u2 = u * u; angP[t] = u2 * u2;
            float v = 1.0f - ct[t]; float v2 = v * v; angM[t] = v2 * v2;
        }

        const float avg  = 0.5f * (Rij + Rik);
        const float base = fij * fik * 0.125f;   // * 2^(1-zeta)
        float rad[5];
#pragma unroll
        for (int rr = 0; rr < 5; ++rr) {
            const float x = avg - Rs[rr];
            rad[rr] = __expf(-ita * x * x) * base;
        }

#pragma unroll
        for (int rr = 0; rr < 5; ++rr) {
#pragma unroll
            for (int t = 0; t < 4; ++t) {
                acc[0 * 20 + rr * 4 + t] += angP[t] * rad[rr];  // lambda = +1
                acc[1 * 20 + rr * 4 + t] += angM[t] * rad[rr];  // lambda = -1
            }
        }
    }

    // block reduction: red[p*128 + tid], then 40 threads finish
#pragma unroll
    for (int p = 0; p < NP; ++p) red[p * NT + tid] = acc[p];
    __syncthreads();

    if (tid < NP) {
        float sum = 0.0f;
        for (int q = 0; q < NT; ++q) sum += red[tid * NT + q];
        out[bi * NP + tid] = sum;
    }
}

extern "C" void kernel_launch(void* const* d_in, const int* in_sizes, int n_in,
                              void* d_out, int out_size, void* d_ws, size_t ws_size,
                              hipStream_t stream) {
    const float* d_cutoff = (const float*)d_in[0];  // (16,32,32)
    const float* d        = (const float*)d_in[1];  // (16,32,32)
    const float* coords   = (const float*)d_in[2];  // (16,32,3)
    float* out            = (float*)d_out;          // (16,32,40)

    const int B = 16, N = 32;
    AngularSymmetry_gfx1250_kernel<<<B * N, 128, 0, stream>>>(d_cutoff, d, coords, out);
}